// Train_56358560858806
// MI455X (gfx1250) — compile-verified
//
#include <hip/hip_runtime.h>
#include <hip/hip_bf16.h>

#define B_TOTAL 64
#define P_TOTAL 24564
#define C_CLS   21
#define T_TRUTH 16

typedef __attribute__((ext_vector_type(2))) float v2f;
typedef __attribute__((ext_vector_type(8))) float v8f;

static __device__ __forceinline__ float smooth_l1(float d) {
    float a = fabsf(d);
    return a < 1.0f ? 0.5f * a * a : a - 0.5f;
}

// ---------------------------------------------------------------------------
// init: zero accumulators
// ---------------------------------------------------------------------------
__global__ void init_kernel(unsigned long long* keys, int* numPos, float* negSum,
                            float* gLossL, float* gLossC) {
    int i = blockIdx.x * blockDim.x + threadIdx.x;
    if (i < B_TOTAL * T_TRUTH) keys[i] = 0ull;
    if (i < B_TOTAL) { numPos[i] = 0; negSum[i] = 0.0f; }
    if (i == 0) { *gLossL = 0.0f; *gLossC = 0.0f; }
}

// ---------------------------------------------------------------------------
// match: per (b,p) best truth; per (b,t) best prior via packed u64 atomicMax
// key = (iou_bits << 32) | ~p  -> max picks max iou, ties pick smallest p
// ---------------------------------------------------------------------------
__global__ void __launch_bounds__(256) match_kernel(
    const float* __restrict__ priors, const float* __restrict__ targets,
    float* __restrict__ bestOv, int* __restrict__ bestIdx,
    unsigned long long* __restrict__ bestPriorKey) {
    int b = blockIdx.y;
    int p = blockIdx.x * blockDim.x + threadIdx.x;

    __shared__ float tg[T_TRUTH * 5];
    __shared__ unsigned long long keys[T_TRUTH];
    if (threadIdx.x < T_TRUTH * 5) tg[threadIdx.x] = targets[b * T_TRUTH * 5 + threadIdx.x];
    if (threadIdx.x < T_TRUTH) keys[threadIdx.x] = 0ull;
    __syncthreads();

    bool valid = p < P_TOTAL;
    int pc = valid ? p : (P_TOTAL - 1);
    float4 pr = ((const float4*)priors)[pc];  // cx cy w h
    float ax1 = pr.x - pr.z * 0.5f, ay1 = pr.y - pr.w * 0.5f;
    float ax2 = pr.x + pr.z * 0.5f, ay2 = pr.y + pr.w * 0.5f;
    float areaP = (ax2 - ax1) * (ay2 - ay1);

    float bestO = -1.0f; int bestT = 0;
#pragma unroll
    for (int t = 0; t < T_TRUTH; t++) {
        float tx1 = tg[t*5+0], ty1 = tg[t*5+1], tx2 = tg[t*5+2], ty2 = tg[t*5+3];
        float lx = fmaxf(tx1, ax1), ly = fmaxf(ty1, ay1);
        float rx = fminf(tx2, ax2), ry = fminf(ty2, ay2);
        float w = fmaxf(rx - lx, 0.0f), h = fmaxf(ry - ly, 0.0f);
        float inter = w * h;
        float areaT = (tx2 - tx1) * (ty2 - ty1);
        float iou = inter / (areaT + areaP - inter);
        if (iou > bestO) { bestO = iou; bestT = t; }
        if (valid) {
            unsigned long long key =
                ((unsigned long long)__float_as_uint(iou) << 32) |
                (unsigned long long)(~(unsigned int)p);
            atomicMax(&keys[t], key);
        }
    }
    if (valid) {
        size_t idx = (size_t)b * P_TOTAL + p;
        bestOv[idx] = bestO;
        bestIdx[idx] = bestT;
    }
    __syncthreads();
    if (threadIdx.x < T_TRUTH)
        atomicMax(&bestPriorKey[b * T_TRUTH + threadIdx.x], keys[threadIdx.x]);
}

// ---------------------------------------------------------------------------
// override: sequential over t (last-wins) to mirror JAX scatter semantics
// ---------------------------------------------------------------------------
__global__ void override_kernel(float* bestOv, int* bestIdx,
                                const unsigned long long* __restrict__ key) {
    int b = blockIdx.x * blockDim.x + threadIdx.x;
    if (b >= B_TOTAL) return;
    for (int t = 0; t < T_TRUTH; t++) {
        unsigned long long k = key[b * T_TRUTH + t];
        unsigned int p = ~(unsigned int)(k & 0xFFFFFFFFull);
        if (p < P_TOTAL) {
            size_t idx = (size_t)b * P_TOTAL + p;
            bestOv[idx] = 2.0f;
            bestIdx[idx] = t;
        }
    }
}

// ---------------------------------------------------------------------------
// fused main pass: logsumexp via V_WMMA_F32_16X16X4_F32 row-sums, loc loss,
// conf loss, positive counting, store negative-loss bits for radix select.
// Each wave32 owns 16 priors; lanes arranged per the f32 A-matrix layout:
//   lane L -> row M = L&15, K pair = {0,1} (lanes 0-15) or {2,3} (lanes 16-31)
// ---------------------------------------------------------------------------
__global__ void __launch_bounds__(256) fused_loss_kernel(
    const float* __restrict__ loc, const float* __restrict__ conf,
    const float* __restrict__ priors, const float* __restrict__ targets,
    const float* __restrict__ bestOv, const int* __restrict__ bestIdx,
    unsigned int* __restrict__ lcmBits, int* __restrict__ numPos,
    float* __restrict__ gLossL, float* __restrict__ gLossC) {
    int b = blockIdx.y;
    int lane = threadIdx.x & 31;
    int wv   = threadIdx.x >> 5;
    int p    = blockIdx.x * 128 + wv * 16 + (lane & 15);
    int row  = p < P_TOTAL ? p : (P_TOTAL - 1);
    const float* crow = conf + ((size_t)b * P_TOTAL + row) * C_CLS;
    __builtin_prefetch(crow, 0, 0);

    int hi = lane >> 4;  // which K pair this lane supplies

    // Load the 12 class logits this lane owns (classes c with ((c>>1)&1)==hi)
    float l[12];
    float m = -INFINITY;
#pragma unroll
    for (int j = 0; j < 6; j++) {
        int c0 = 4 * j + 2 * hi;
        float a0 = (c0     < C_CLS) ? crow[c0]     : -INFINITY;
        float a1 = (c0 + 1 < C_CLS) ? crow[c0 + 1] : -INFINITY;
        l[2*j]   = a0;
        l[2*j+1] = a1;
        m = fmaxf(m, fmaxf(a0, a1));
    }
    // full per-prior max (the other 12 classes live in lane ^ 16)
    m = fmaxf(m, __shfl_xor(m, 16));

    // Row-sum of exp via 6 accumulating f32 WMMAs against an all-ones B.
    v8f acc = {};
    v2f ones; ones.x = 1.0f; ones.y = 1.0f;
#pragma unroll
    for (int j = 0; j < 6; j++) {
        v2f a;
        a.x = expf(l[2*j]   - m);   // expf(-inf - m) == 0 handles padding
        a.y = expf(l[2*j+1] - m);
        acc = __builtin_amdgcn_wmma_f32_16x16x4_f32(
            false, a, false, ones, (short)0, acc, false, false);
    }

    // D[M][N] = rowsum[M] for every N. VGPR j holds M=j (lanes 0-15) or
    // M=j+8 (lanes 16-31). Pull rowsum[lane&15] via select ladder + bpermute.
    int i7 = lane & 7;
    float x = acc[0];
    x = (i7 == 1) ? acc[1] : x;
    x = (i7 == 2) ? acc[2] : x;
    x = (i7 == 3) ? acc[3] : x;
    x = (i7 == 4) ? acc[4] : x;
    x = (i7 == 5) ? acc[5] : x;
    x = (i7 == 6) ? acc[6] : x;
    x = (i7 == 7) ? acc[7] : x;
    int src = (lane & 7) | (((lane >> 3) & 1) << 4);
    float s = __shfl(x, src);
    float lse = m + logf(s);

    bool active = (lane < 16) && (p < P_TOTAL);
    float lossL_loc = 0.0f, lossC_loc = 0.0f;
    int np = 0;
    if (active) {
        size_t idx = (size_t)b * P_TOTAL + p;
        float ov = bestOv[idx];
        int   t  = bestIdx[idx];
        bool pos = ov >= 0.5f;
        const float* tr = targets + ((size_t)b * T_TRUTH + t) * 5;
        int conf_t = pos ? ((int)tr[4] + 1) : 0;
        float tgt = crow[conf_t];
        float lca = lse - tgt;
        lcmBits[idx] = pos ? 0u : __float_as_uint(fmaxf(lca, 0.0f));
        if (pos) {
            np = 1;
            lossC_loc = lca;
            float4 pr = ((const float4*)priors)[p];
            float mx1 = tr[0], my1 = tr[1], mx2 = tr[2], my2 = tr[3];
            float gcx = ((mx1 + mx2) * 0.5f - pr.x) / (0.1f * pr.z);
            float gcy = ((my1 + my2) * 0.5f - pr.y) / (0.1f * pr.w);
            float gw  = logf((mx2 - mx1) / pr.z) / 0.2f;
            float gh  = logf((my2 - my1) / pr.w) / 0.2f;
            const float* ld = loc + idx * 4;
            lossL_loc = smooth_l1(ld[0] - gcx) + smooth_l1(ld[1] - gcy) +
                        smooth_l1(ld[2] - gw)  + smooth_l1(ld[3] - gh);
        }
    }

    // block reduction
    __shared__ float sL, sC;
    __shared__ int sN;
    if (threadIdx.x == 0) { sL = 0.0f; sC = 0.0f; sN = 0; }
    __syncthreads();
#pragma unroll
    for (int off = 16; off > 0; off >>= 1) {
        lossL_loc += __shfl_down(lossL_loc, off);
        lossC_loc += __shfl_down(lossC_loc, off);
        np        += __shfl_down(np, off);
    }
    if (lane == 0) {
        atomicAdd(&sL, lossL_loc);
        atomicAdd(&sC, lossC_loc);
        atomicAdd(&sN, np);
    }
    __syncthreads();
    if (threadIdx.x == 0) {
        atomicAdd(gLossL, sL);
        atomicAdd(gLossC, sC);
        atomicAdd(&numPos[b], sN);
    }
}

// ---------------------------------------------------------------------------
// topk: per-batch MSB-first 4x8-bit radix select; sum of K largest values.
// Values are non-negative floats -> uint bit pattern is order-preserving.
// ---------------------------------------------------------------------------
__global__ void __launch_bounds__(256) topk_kernel(
    const unsigned int* __restrict__ lcmBits, const int* __restrict__ numPos,
    float* __restrict__ negSum) {
    int b = blockIdx.x;
    const unsigned int* Bb = lcmBits + (size_t)b * P_TOTAL;

    int K = numPos[b] * 3;
    if (K > P_TOTAL - 1) K = P_TOTAL - 1;
    if (K <= 0) {
        if (threadIdx.x == 0) negSum[b] = 0.0f;
        return;
    }

    __shared__ unsigned int hist[256];
    __shared__ unsigned int sPrefix;
    __shared__ int sK;
    if (threadIdx.x == 0) { sPrefix = 0u; sK = K; }
    __syncthreads();

    for (int pass = 0; pass < 4; pass++) {
        hist[threadIdx.x] = 0u;
        __syncthreads();
        unsigned int prefix = sPrefix;
        int shift = 24 - 8 * pass;
        for (int p = threadIdx.x; p < P_TOTAL; p += 256) {
            unsigned int bits = Bb[p];
            bool ok = (pass == 0) || ((bits >> (shift + 8)) == prefix);
            if (ok) atomicAdd(&hist[(bits >> shift) & 0xFFu], 1u);
        }
        __syncthreads();
        if (threadIdx.x == 0) {
            int krem = sK;
            unsigned int cum = 0;
            int d = 0;
            for (int dd = 255; dd >= 0; dd--) {
                unsigned int c = hist[dd];
                if (cum + c >= (unsigned int)krem) { d = dd; break; }
                cum += c;
            }
            sPrefix = (prefix << 8) | (unsigned int)d;
            sK = krem - (int)cum;
        }
        __syncthreads();
    }

    unsigned int thr = sPrefix;
    int kAt = sK;  // elements equal to thr still to take
    float sum = 0.0f;
    for (int p = threadIdx.x; p < P_TOTAL; p += 256) {
        unsigned int bits = Bb[p];
        if (bits > thr) sum += __uint_as_float(bits);
    }
#pragma unroll
    for (int off = 16; off > 0; off >>= 1) sum += __shfl_down(sum, off);
    __shared__ float wsum[8];
    int lane = threadIdx.x & 31, wv = threadIdx.x >> 5;
    if (lane == 0) wsum[wv] = sum;
    __syncthreads();
    if (threadIdx.x == 0) {
        float t = 0.0f;
        for (int i = 0; i < 8; i++) t += wsum[i];
        negSum[b] = t + (float)kAt * __uint_as_float(thr);
    }
}

// ---------------------------------------------------------------------------
// finalize
// ---------------------------------------------------------------------------
__global__ void finalize_kernel(const int* __restrict__ numPos,
                                const float* __restrict__ negSum,
                                const float* __restrict__ gLossL,
                                const float* __restrict__ gLossC,
                                float* __restrict__ out) {
    if (threadIdx.x == 0 && blockIdx.x == 0) {
        int N = 0; float neg = 0.0f;
        for (int b = 0; b < B_TOTAL; b++) { N += numPos[b]; neg += negSum[b]; }
        float lossL = *gLossL;
        float lossC = *gLossC + neg;
        float fN = (float)N;
        float a = lossL / fN;
        float c = lossC / fN;
        out[0] = a; out[1] = c; out[2] = a + c;
    }
}

extern "C" void kernel_launch(void* const* d_in, const int* in_sizes, int n_in,
                              void* d_out, int out_size, void* d_ws, size_t ws_size,
                              hipStream_t stream) {
    const float* loc     = (const float*)d_in[0];
    const float* conf    = (const float*)d_in[1];
    const float* priors  = (const float*)d_in[2];
    const float* targets = (const float*)d_in[3];
    float* out = (float*)d_out;

    const size_t BP = (size_t)B_TOTAL * P_TOTAL;
    char* w = (char*)d_ws;
    unsigned long long* keys = (unsigned long long*)w;            w += (size_t)B_TOTAL * T_TRUTH * 8;
    float*        bestOv  = (float*)w;                            w += BP * 4;
    int*          bestIdx = (int*)w;                              w += BP * 4;
    unsigned int* lcmBits = (unsigned int*)w;                     w += BP * 4;
    int*          numPos  = (int*)w;                              w += B_TOTAL * 4;
    float*        negSum  = (float*)w;                            w += B_TOTAL * 4;
    float*        gLossL  = (float*)w;                            w += 8;
    float*        gLossC  = (float*)w;

    init_kernel<<<4, 256, 0, stream>>>(keys, numPos, negSum, gLossL, gLossC);

    dim3 gm((P_TOTAL + 255) / 256, B_TOTAL);
    match_kernel<<<gm, 256, 0, stream>>>(priors, targets, bestOv, bestIdx, keys);

    override_kernel<<<1, 64, 0, stream>>>(bestOv, bestIdx, keys);

    dim3 gf((P_TOTAL + 127) / 128, B_TOTAL);
    fused_loss_kernel<<<gf, 256, 0, stream>>>(loc, conf, priors, targets,
                                              bestOv, bestIdx, lcmBits,
                                              numPos, gLossL, gLossC);

    topk_kernel<<<B_TOTAL, 256, 0, stream>>>(lcmBits, numPos, negSum);

    finalize_kernel<<<1, 1, 0, stream>>>(numPos, negSum, gLossL, gLossC, out);
}